// InTrackAttentionLayer_44006234915248
// MI455X (gfx1250) — compile-verified
//
#include <hip/hip_runtime.h>
#include <hip/hip_bf16.h>
#include <math.h>

#define N_TRACKS 8192
#define TRACK_LEN 32
#define D_IN 256
#define D_OUT 256
#define NHEADS 4
#define HDIM 64
#define QKV_COLS 768

// LDS pitches (floats). All even -> float2 (8B) alignment holds everywhere.
#define PITCHH 260   // h / ctx buffer: 32 x 260
#define PITCHQ 772   // qkv buffer:     32 x 772
#define PITCHS 34    // scores:     4 x 32 x 34
#define SMEM_FLOATS (32 * PITCHH + 32 * PITCHQ + NHEADS * 32 * PITCHS)
#define SMEM_BYTES (SMEM_FLOATS * 4)

typedef __attribute__((ext_vector_type(2))) float v2f;
typedef __attribute__((ext_vector_type(8))) float v8f;

static __device__ __forceinline__ v8f wmma_f32(v2f a, v2f b, v8f c) {
  // D(16x16,f32) = A(16x4,f32) * B(4x16,f32) + C
  return __builtin_amdgcn_wmma_f32_16x16x4_f32(
      false, a, false, b, (short)0, c, false, false);
}

__global__ void __launch_bounds__(256)
track_attn_kernel(const float* __restrict__ values,
                  const float* __restrict__ ln_gamma,
                  const float* __restrict__ ln_beta,
                  const float* __restrict__ W_qkv,
                  const float* __restrict__ b_qkv,
                  const float* __restrict__ W_out,
                  const float* __restrict__ b_out,
                  float* __restrict__ out) {
  extern __shared__ float smem[];
  float* sH = smem;                       // 32 x PITCHH   (h, later reused as ctx)
  float* sQKV = smem + 32 * PITCHH;       // 32 x PITCHQ   (cols: [Q 0..255][K 256..511][V 512..767])
  float* sS = sQKV + 32 * PITCHQ;         // 4 heads x 32 x PITCHS

  const int tid = threadIdx.x;
  const int lane = tid & 31;
  const int wave = tid >> 5;
  const int track = blockIdx.x;

  const int nlo = lane & 15;          // column within 16-wide tile / A row
  const int koff = (lane >> 4) * 2;   // K sub-offset for A/B fragments
  const int drow = (lane >> 4) * 8;   // D-tile row base for this half-wave

  // ---------------- LayerNorm: 8 lanes per token, 32 elems per lane ----------------
  {
    const int token = tid >> 3;   // 0..31
    const int seg = tid & 7;      // 0..7
    const int cbase = seg * 32;
    const float4* v4 = (const float4*)(values + ((size_t)track * TRACK_LEN + token) * D_IN + cbase);
    float4 v[8];
    float s = 0.f, ss = 0.f;
#pragma unroll
    for (int i = 0; i < 8; ++i) {
      v[i] = v4[i];
      s += v[i].x + v[i].y + v[i].z + v[i].w;
      ss += v[i].x * v[i].x + v[i].y * v[i].y + v[i].z * v[i].z + v[i].w * v[i].w;
    }
#pragma unroll
    for (int m = 1; m <= 4; m <<= 1) {
      s += __shfl_xor(s, m, 32);
      ss += __shfl_xor(ss, m, 32);
    }
    const float mu = s * (1.f / 256.f);
    const float var = ss * (1.f / 256.f) - mu * mu;
    const float rs = rsqrtf(var + 1e-6f);
    const float4* g4 = (const float4*)(ln_gamma + cbase);
    const float4* b4 = (const float4*)(ln_beta + cbase);
    float4* h4 = (float4*)(sH + token * PITCHH + cbase);
#pragma unroll
    for (int i = 0; i < 8; ++i) {
      float4 g = g4[i], bb = b4[i], hv;
      hv.x = (v[i].x - mu) * rs * g.x + bb.x;
      hv.y = (v[i].y - mu) * rs * g.y + bb.y;
      hv.z = (v[i].z - mu) * rs * g.z + bb.z;
      hv.w = (v[i].w - mu) * rs * g.w + bb.w;
      h4[i] = hv;
    }
  }
  __syncthreads();

  // ---------------- QKV GEMM: (32x256) @ (256x768) + b ----------------
  {
    v8f acc[6][2];
#pragma unroll
    for (int j = 0; j < 6; ++j) {
      const int nt = wave + j * 8;  // n-tile 0..47
      const float bv = b_qkv[nt * 16 + nlo];
#pragma unroll
      for (int mt = 0; mt < 2; ++mt)
#pragma unroll
        for (int r = 0; r < 8; ++r) acc[j][mt][r] = bv;
    }
    for (int k0 = 0; k0 < D_IN; k0 += 4) {
      const v2f a0 = *(const v2f*)(sH + nlo * PITCHH + k0 + koff);
      const v2f a1 = *(const v2f*)(sH + (16 + nlo) * PITCHH + k0 + koff);
#pragma unroll
      for (int j = 0; j < 6; ++j) {
        const int n = (wave + j * 8) * 16 + nlo;
        v2f b;
        b.x = W_qkv[(size_t)(k0 + koff) * QKV_COLS + n];
        b.y = W_qkv[(size_t)(k0 + koff + 1) * QKV_COLS + n];
        acc[j][0] = wmma_f32(a0, b, acc[j][0]);
        acc[j][1] = wmma_f32(a1, b, acc[j][1]);
      }
    }
#pragma unroll
    for (int j = 0; j < 6; ++j) {
      const int ncol = (wave + j * 8) * 16 + nlo;
#pragma unroll
      for (int mt = 0; mt < 2; ++mt)
#pragma unroll
        for (int r = 0; r < 8; ++r)
          sQKV[(mt * 16 + drow + r) * PITCHQ + ncol] = acc[j][mt][r];
    }
  }
  __syncthreads();

  // ---------------- scores = Q @ K^T / 8 : per head 32x32, 16 tiles over 8 waves ---
  {
#pragma unroll
    for (int tt = 0; tt < 2; ++tt) {
      const int t = wave * 2 + tt;
      const int head = t >> 2;
      const int mt = (t >> 1) & 1;
      const int nt = t & 1;
      v8f acc = {};
      const float* qrow = sQKV + (mt * 16 + nlo) * PITCHQ + head * HDIM;         // A: Q row m
      const float* krow = sQKV + (nt * 16 + nlo) * PITCHQ + D_OUT + head * HDIM; // B: K^T -> K row n
      for (int k0 = 0; k0 < HDIM; k0 += 4) {
        const v2f a = *(const v2f*)(qrow + k0 + koff);
        const v2f b = *(const v2f*)(krow + k0 + koff);
        acc = wmma_f32(a, b, acc);
      }
      float* srow = sS + head * 32 * PITCHS + nt * 16 + nlo;
#pragma unroll
      for (int r = 0; r < 8; ++r)
        srow[(mt * 16 + drow + r) * PITCHS] = acc[r] * 0.125f;
    }
  }
  __syncthreads();

  // ---------------- softmax: 128 rows (4 heads x 32), one row per thread ----------
  if (tid < NHEADS * 32) {
    float* p = sS + (tid >> 5) * 32 * PITCHS + (tid & 31) * PITCHS;
    float mx = -3.402823466e38f;
#pragma unroll
    for (int j = 0; j < 32; ++j) mx = fmaxf(mx, p[j]);
    float sum = 0.f;
#pragma unroll
    for (int j = 0; j < 32; ++j) {
      const float e = __expf(p[j] - mx);
      p[j] = e;
      sum += e;
    }
    const float inv = 1.f / sum;
#pragma unroll
    for (int j = 0; j < 32; ++j) p[j] *= inv;
  }
  __syncthreads();

  // ---------------- ctx = P @ V : per head 32x64, 32 tiles over 8 waves -----------
  float* sC = sH;  // reuse h buffer for ctx (32 x 256 in 32 x PITCHH)
  {
#pragma unroll
    for (int tt = 0; tt < 4; ++tt) {
      const int t = wave * 4 + tt;
      const int head = t >> 3;
      const int mt = (t >> 2) & 1;
      const int nt = t & 3;
      v8f acc = {};
      const float* arow = sS + head * 32 * PITCHS + (mt * 16 + nlo) * PITCHS;   // A: P row m
      const float* bcol = sQKV + 2 * D_OUT + head * HDIM + nt * 16 + nlo;       // B: V[k][n]
      for (int k0 = 0; k0 < TRACK_LEN; k0 += 4) {
        const v2f a = *(const v2f*)(arow + k0 + koff);
        v2f b;
        b.x = bcol[(k0 + koff) * PITCHQ];
        b.y = bcol[(k0 + koff + 1) * PITCHQ];
        acc = wmma_f32(a, b, acc);
      }
      const int ncol = head * HDIM + nt * 16 + nlo;
#pragma unroll
      for (int r = 0; r < 8; ++r)
        sC[(mt * 16 + drow + r) * PITCHH + ncol] = acc[r];
    }
  }
  __syncthreads();

  // ---------------- out = ctx @ W_out + b_out : (32x256)@(256x256) ----------------
  {
    v8f acc[2][2];
#pragma unroll
    for (int j = 0; j < 2; ++j) {
      const float bv = b_out[(wave + j * 8) * 16 + nlo];
#pragma unroll
      for (int mt = 0; mt < 2; ++mt)
#pragma unroll
        for (int r = 0; r < 8; ++r) acc[j][mt][r] = bv;
    }
    for (int k0 = 0; k0 < D_OUT; k0 += 4) {
      const v2f a0 = *(const v2f*)(sC + nlo * PITCHH + k0 + koff);
      const v2f a1 = *(const v2f*)(sC + (16 + nlo) * PITCHH + k0 + koff);
#pragma unroll
      for (int j = 0; j < 2; ++j) {
        const int n = (wave + j * 8) * 16 + nlo;
        v2f b;
        b.x = W_out[(size_t)(k0 + koff) * D_OUT + n];
        b.y = W_out[(size_t)(k0 + koff + 1) * D_OUT + n];
        acc[j][0] = wmma_f32(a0, b, acc[j][0]);
        acc[j][1] = wmma_f32(a1, b, acc[j][1]);
      }
    }
#pragma unroll
    for (int j = 0; j < 2; ++j) {
      const int n = (wave + j * 8) * 16 + nlo;
#pragma unroll
      for (int mt = 0; mt < 2; ++mt)
#pragma unroll
        for (int r = 0; r < 8; ++r) {
          const size_t row = (size_t)track * TRACK_LEN + mt * 16 + drow + r;
          out[row * D_OUT + n] = acc[j][mt][r];
        }
    }
  }
}

extern "C" void kernel_launch(void* const* d_in, const int* in_sizes, int n_in,
                              void* d_out, int out_size, void* d_ws, size_t ws_size,
                              hipStream_t stream) {
  const float* values = (const float*)d_in[0];
  // d_in[1] = indices (int64) — unused by the reference computation
  const float* ln_gamma = (const float*)d_in[2];
  const float* ln_beta = (const float*)d_in[3];
  const float* W_qkv = (const float*)d_in[4];
  const float* b_qkv = (const float*)d_in[5];
  const float* W_out = (const float*)d_in[6];
  const float* b_out = (const float*)d_in[7];
  float* outp = (float*)d_out;

  (void)in_sizes; (void)n_in; (void)out_size; (void)d_ws; (void)ws_size;

  hipFuncSetAttribute((const void*)track_attn_kernel,
                      hipFuncAttributeMaxDynamicSharedMemorySize, SMEM_BYTES);
  track_attn_kernel<<<dim3(N_TRACKS), dim3(256), SMEM_BYTES, stream>>>(
      values, ln_gamma, ln_beta, W_qkv, b_qkv, W_out, b_out, outp);
}